// TtMambaSSM_58351425683716
// MI455X (gfx1250) — compile-verified
//
#include <hip/hip_runtime.h>

#define U_DIM 128
#define DIN   10240
#define RNK   320
#define NST   32
#define NCAT  384          // RNK + 2*NST
#define SPLITK 32

typedef __attribute__((ext_vector_type(2))) float v2f;
typedef __attribute__((ext_vector_type(8))) float v8f;

// ---------------- zero split-K accumulator ----------------
__global__ void zero_ws_kernel(float* __restrict__ p, int n) {
    int i = blockIdx.x * blockDim.x + threadIdx.x;
    if (i < n) p[i] = 0.0f;
}

// ---------------- GEMM 1: x @ [W_delta | W_B | W_C] -> Tcat[128,384] ----------------
// fp32 WMMA 16x16x4, split-K=32 with f32 atomics. K-loop unrolled x8 so 24
// loads are in flight per wmma burst instead of load->wait->wmma per K step.
__global__ __launch_bounds__(32) void gemm_cat_kernel(
    const float* __restrict__ x,
    const float* __restrict__ Wd,
    const float* __restrict__ Wb,
    const float* __restrict__ Wc,
    float* __restrict__ Tcat) {
    const int lane = threadIdx.x & 31;
    int bid = blockIdx.x;
    const int mt = bid % (U_DIM / 16); bid /= (U_DIM / 16);
    const int nt = bid % (NCAT / 16);  bid /= (NCAT / 16);
    const int ks = bid;                              // 0..SPLITK-1
    const int kbeg = ks * (DIN / SPLITK);            // 320 K per wave

    const int m     = mt * 16 + (lane & 15);
    const int ncol  = nt * 16 + (lane & 15);
    const int khalf = (lane >> 4) * 2;               // ISA 16x4 A layout: k = v + 2*(lane>=16)

    // select source column of the concatenated [W_delta | W_B | W_C]
    const float* bptr; int bstride;
    if (ncol < RNK)            { bptr = Wd + ncol;               bstride = RNK; }
    else if (ncol < RNK + NST) { bptr = Wb + (ncol - RNK);       bstride = NST; }
    else                       { bptr = Wc + (ncol - RNK - NST); bstride = NST; }

    const float* __restrict__ arow = x + (size_t)m * DIN;

    v8f acc = {};
#pragma unroll 8
    for (int kk = 0; kk < DIN / SPLITK; kk += 4) {
        const int k0 = kbeg + kk + khalf;
        v2f a; a.x = arow[k0]; a.y = arow[k0 + 1];
        v2f b; b.x = bptr[(size_t)k0 * bstride];
               b.y = bptr[(size_t)(k0 + 1) * bstride];
        acc = __builtin_amdgcn_wmma_f32_16x16x4_f32(false, a, false, b,
                                                    (short)0, acc, false, false);
    }

    const int rbase = mt * 16 + 8 * (lane >> 4);     // C/D layout: row = i + 8*(lane>=16)
#pragma unroll
    for (int i = 0; i < 8; ++i)
        atomicAdd(&Tcat[(size_t)(rbase + i) * NCAT + ncol], acc[i]);
}

// ---------------- GEMM 2: Tcat[:, :320] @ W_dt + b_dt -> softplus -> delta[128,10240] ----------------
__global__ __launch_bounds__(32) void gemm_dt_kernel(
    const float* __restrict__ Tcat,
    const float* __restrict__ Wdt,
    const float* __restrict__ bdt,
    float* __restrict__ delta) {
    const int lane = threadIdx.x & 31;
    const int mt = blockIdx.x % (U_DIM / 16);
    const int nt = blockIdx.x / (U_DIM / 16);        // 0..639
    const int m     = mt * 16 + (lane & 15);
    const int n     = nt * 16 + (lane & 15);
    const int khalf = (lane >> 4) * 2;

    const float* __restrict__ trow = Tcat + (size_t)m * NCAT;

    v8f acc = {};
#pragma unroll 8
    for (int k = 0; k < RNK; k += 4) {
        const int k0 = k + khalf;
        v2f a; a.x = trow[k0];
               a.y = trow[k0 + 1];
        v2f b; b.x = Wdt[(size_t)k0 * DIN + n];
               b.y = Wdt[(size_t)(k0 + 1) * DIN + n];
        acc = __builtin_amdgcn_wmma_f32_16x16x4_f32(false, a, false, b,
                                                    (short)0, acc, false, false);
    }

    const float bias  = bdt[n];
    const int   rbase = mt * 16 + 8 * (lane >> 4);
#pragma unroll
    for (int i = 0; i < 8; ++i) {
        const float v  = acc[i] + bias;
        const float sp = (v > 20.0f) ? v : __logf(1.0f + __expf(v));   // softplus
        delta[(size_t)(rbase + i) * DIN + n] = sp;
    }
}

// ---------------- fused state update + output (memory-bound phase) ----------------
// One wave per (u, 128 d's); lane = n. h read/write fully coalesced (128B/instr).
// ~336 MB of h traffic dominates the whole problem: ~16 us at 23.3 TB/s.
__global__ __launch_bounds__(256) void ssm_stream_kernel(
    const float* __restrict__ x,
    const float* __restrict__ h,
    const float* __restrict__ A,
    const float* __restrict__ Dv,
    const float* __restrict__ delta,
    const float* __restrict__ Tcat,
    float* __restrict__ y,
    float* __restrict__ hnew) {
    const int lane = threadIdx.x & 31;
    const int wid  = blockIdx.x * (blockDim.x >> 5) + (threadIdx.x >> 5);
    const int WPU  = DIN / 128;                      // 80 waves per u
    const int u    = wid / WPU;
    const int d0   = (wid % WPU) * 128;

    const float Bv = Tcat[(size_t)u * NCAT + RNK + lane];        // B[u, lane]
    const float Cv = Tcat[(size_t)u * NCAT + RNK + NST + lane];  // C[u, lane]

    float yreg = 0.0f;
#pragma unroll 2
    for (int d = d0; d < d0 + 128; ++d) {
        const float dl = delta[(size_t)u * DIN + d];             // wave-uniform
        const float xv = x[(size_t)u * DIN + d];                 // wave-uniform
        const float Av = A[(size_t)d * NST + lane];              // L2-resident (1.3 MB)
        const size_t hidx = ((size_t)u * DIN + d) * NST + lane;

        const float hv   = h[hidx];
        const float abar = __expf(dl * Av);
        const float hn   = fmaf(abar, hv, dl * xv * Bv);
        hnew[hidx] = hn;

        float p = hn * Cv;                                       // reduce over n
        p += __shfl_xor(p, 16, 32);
        p += __shfl_xor(p,  8, 32);
        p += __shfl_xor(p,  4, 32);
        p += __shfl_xor(p,  2, 32);
        p += __shfl_xor(p,  1, 32);
        const float ysum = p + Dv[d] * xv;

        if (lane == (d & 31)) yreg = ysum;                       // bank per-lane
        if ((d & 31) == 31)                                      // coalesced y store
            y[(size_t)u * DIN + (d - 31) + lane] = yreg;
    }
}

extern "C" void kernel_launch(void* const* d_in, const int* in_sizes, int n_in,
                              void* d_out, int out_size, void* d_ws, size_t ws_size,
                              hipStream_t stream) {
    const float* x   = (const float*)d_in[0];
    const float* h   = (const float*)d_in[1];
    const float* Wd  = (const float*)d_in[2];
    const float* Wdt = (const float*)d_in[3];
    const float* bdt = (const float*)d_in[4];
    const float* Wb  = (const float*)d_in[5];
    const float* Wc  = (const float*)d_in[6];
    const float* A   = (const float*)d_in[7];
    const float* Dv  = (const float*)d_in[8];

    float* Tcat  = (float*)d_ws;                       // [128, 384]
    float* delta = Tcat + (size_t)U_DIM * NCAT;        // [128, 10240]

    float* y    = (float*)d_out;                       // [128, 10240]
    float* hnew = y + (size_t)U_DIM * DIN;             // [128, 10240, 32]

    const int ztot = U_DIM * NCAT;
    zero_ws_kernel<<<(ztot + 255) / 256, 256, 0, stream>>>(Tcat, ztot);

    gemm_cat_kernel<<<(U_DIM / 16) * (NCAT / 16) * SPLITK, 32, 0, stream>>>(
        x, Wd, Wb, Wc, Tcat);

    gemm_dt_kernel<<<(U_DIM / 16) * (DIN / 16), 32, 0, stream>>>(
        Tcat, Wdt, bdt, delta);

    const int nwaves = U_DIM * (DIN / 128);            // 10240 waves
    ssm_stream_kernel<<<nwaves / 8, 256, 0, stream>>>(
        x, h, A, Dv, delta, Tcat, y, hnew);
}